// GraphSAGEAutoencoder_18081812316503
// MI455X (gfx1250) — compile-verified
//
#include <hip/hip_runtime.h>

#define NN   50000
#define NE   800000
#define DIN  128
#define DH   64
#define DOUTC 64

typedef __attribute__((ext_vector_type(2))) float v2f;
typedef __attribute__((ext_vector_type(8))) float v8f;

// ---------------- init: zero agg1, agg2, deg ----------------
__global__ void zero_kernel(float* p, int n) {
  int i = blockIdx.x * blockDim.x + threadIdx.x;
  int stride = gridDim.x * blockDim.x;
  for (; i < n; i += stride) p[i] = 0.0f;
}

// ---------------- degree: one thread per edge ----------------
__global__ void degree_kernel(const int* __restrict__ ei, float* __restrict__ deg) {
  int e = blockIdx.x * blockDim.x + threadIdx.x;
  if (e < NE) {
    int dst = ei[NE + e];
    atomicAdd(&deg[dst], 1.0f);
  }
}

// ---------------- scatter: F/4 lanes per edge, float4 gather + 4 atomics ----------------
template <int F>
__global__ void scatter_kernel(const int* __restrict__ ei, const float* __restrict__ x,
                               float* __restrict__ agg) {
  constexpr int LPE = F / 4;
  int t = blockIdx.x * blockDim.x + threadIdx.x;
  int e = t / LPE;
  int f4 = t - e * LPE;
  if (e < NE) {
    int src = ei[e];
    int dst = ei[NE + e];
    float4 v = ((const float4*)(x + (long long)src * F))[f4];
    float* o = agg + (long long)dst * F + f4 * 4;
    atomicAdd(o + 0, v.x);
    atomicAdd(o + 1, v.y);
    atomicAdd(o + 2, v.z);
    atomicAdd(o + 3, v.w);
  }
}

// ---------------- fused dual-GEMM via V_WMMA_F32_16X16X4_F32 ----------------
// out[16-row tile, NOUT] = act( scale(A1) @ W1 + bias + (DUAL ? A2 @ W2 : 0) )
// One block = NOUT/16 waves; wave w owns 16-col tile w. A tiles staged in LDS.
template <int K, int NOUT, bool DUAL, bool RELU, bool SCALE>
__global__ __launch_bounds__(NOUT * 2) void sage_gemm_kernel(
    const float* __restrict__ A1, const float* __restrict__ deg,
    const float* __restrict__ A2, const float* __restrict__ W1,
    const float* __restrict__ W2, const float* __restrict__ bias,
    float* __restrict__ out) {
  constexpr int LDK = K + 4;  // pad so 16-lane column reads hit distinct banks
  __shared__ float A1s[16 * LDK];
  __shared__ float A2s[DUAL ? 16 * LDK : 4];

  const int tid = threadIdx.x;
  constexpr int NTHREADS = NOUT * 2;
  const int rowbase = blockIdx.x * 16;

  // Stage A tiles (16 x K) into LDS with coalesced float4 loads.
  constexpr int K4 = K / 4;
  for (int idx = tid; idx < 16 * K4; idx += NTHREADS) {
    int r = idx / K4;
    int c4 = idx - r * K4;
    float s = 1.0f;
    if (SCALE) s = 1.0f / fmaxf(deg[rowbase + r], 1.0f);
    float4 v = ((const float4*)(A1 + (long long)(rowbase + r) * K))[c4];
    float* d1 = &A1s[r * LDK + c4 * 4];
    d1[0] = v.x * s; d1[1] = v.y * s; d1[2] = v.z * s; d1[3] = v.w * s;
    if (DUAL) {
      float4 u = ((const float4*)(A2 + (long long)(rowbase + r) * K))[c4];
      float* d2 = &A2s[r * LDK + c4 * 4];
      d2[0] = u.x; d2[1] = u.y; d2[2] = u.z; d2[3] = u.w;
    }
  }
  __syncthreads();

  const int wave = tid >> 5;       // 16-col tile index
  const int lane = tid & 31;
  const int half = lane >> 4;      // K sub-pair select (A/B), row+8 select (C/D)
  const int lid  = lane & 15;      // A row / B,C,D column-in-tile
  const int col  = wave * 16 + lid;

  v8f c = {};
#pragma unroll 4
  for (int k0 = 0; k0 < K; k0 += 4) {
    const int ka = k0 + 2 * half;
    v2f a1, b1;
    a1.x = A1s[lid * LDK + ka];
    a1.y = A1s[lid * LDK + ka + 1];
    b1.x = W1[ka * NOUT + col];
    b1.y = W1[(ka + 1) * NOUT + col];
    c = __builtin_amdgcn_wmma_f32_16x16x4_f32(false, a1, false, b1, (short)0, c,
                                              false, false);
    if (DUAL) {
      v2f a2, b2;
      a2.x = A2s[lid * LDK + ka];
      a2.y = A2s[lid * LDK + ka + 1];
      b2.x = W2[ka * NOUT + col];
      b2.y = W2[(ka + 1) * NOUT + col];
      c = __builtin_amdgcn_wmma_f32_16x16x4_f32(false, a2, false, b2, (short)0, c,
                                                false, false);
    }
  }

  const float bv = bias[col];
#pragma unroll
  for (int g = 0; g < 8; ++g) {
    float v = c[g] + bv;
    if (RELU) v = fmaxf(v, 0.0f);
    out[(long long)(rowbase + g + 8 * half) * NOUT + col] = v;
  }
}

// ---------------- launcher ----------------
extern "C" void kernel_launch(void* const* d_in, const int* in_sizes, int n_in,
                              void* d_out, int out_size, void* d_ws, size_t ws_size,
                              hipStream_t stream) {
  const float* x   = (const float*)d_in[0];   // [N, 128]
  const int*   ei  = (const int*)d_in[1];     // [2, E]
  const float* W1l = (const float*)d_in[2];   // [128, 64]
  const float* b1  = (const float*)d_in[3];   // [64]
  const float* W1r = (const float*)d_in[4];   // [128, 64]
  const float* W2l = (const float*)d_in[5];   // [64, 64]
  const float* b2  = (const float*)d_in[6];   // [64]
  const float* W2r = (const float*)d_in[7];   // [64, 64]
  const float* Wd  = (const float*)d_in[8];   // [64, 128]
  const float* bd  = (const float*)d_in[9];   // [128]

  float* outz = (float*)d_out;                // z: [N, 64]
  float* outx = outz + (long long)NN * DOUTC; // x_hat: [N, 128]

  float* ws   = (float*)d_ws;
  float* agg1 = ws;                           // N*128
  float* agg2 = agg1 + (long long)NN * DIN;   // N*64
  float* deg  = agg2 + (long long)NN * DH;    // N
  float* h    = deg + NN;                     // N*64

  // 1) zero accumulators (agg1, agg2, deg are contiguous in ws)
  {
    int nz = NN * (DIN + DH + 1);
    zero_kernel<<<4096, 256, 0, stream>>>(ws, nz);
  }
  // 2) degrees
  degree_kernel<<<(NE + 255) / 256, 256, 0, stream>>>(ei, deg);
  // 3) scatter x -> agg1 (32 lanes/edge)
  scatter_kernel<DIN><<<(NE * (DIN / 4) + 255) / 256, 256, 0, stream>>>(ei, x, agg1);
  // 4) h = relu((agg1/deg) @ W1l + b1 + x @ W1r)
  sage_gemm_kernel<DIN, DH, true, true, true>
      <<<NN / 16, DH * 2, 0, stream>>>(agg1, deg, x, W1l, W1r, b1, h);
  // 5) scatter h -> agg2 (16 lanes/edge)
  scatter_kernel<DH><<<(NE * (DH / 4) + 255) / 256, 256, 0, stream>>>(ei, h, agg2);
  // 6) z = (agg2/deg) @ W2l + b2 + h @ W2r   (written directly to d_out)
  sage_gemm_kernel<DH, DOUTC, true, false, true>
      <<<NN / 16, DOUTC * 2, 0, stream>>>(agg2, deg, h, W2l, W2r, b2, outz);
  // 7) x_hat = z @ Wd + bd
  sage_gemm_kernel<DOUTC, DIN, false, false, false>
      <<<NN / 16, DIN * 2, 0, stream>>>(outz, deg, nullptr, Wd, nullptr, bd, outx);
}